// DotProductPredictor_40724879901345
// MI455X (gfx1250) — compile-verified
//
#include <hip/hip_runtime.h>

// Per-edge dot product (DGL u_dot_v): score[e] = dot(h[src[e]], h[dst[e]])
// E = 640k, D = 128, f32. Gather-bound out of L2 (h table = 5.1 MB << 192 MB L2).
// Strategy: 16 edges per wave; C(16x16) = Hu(16x128) x Hv^T(128x16) via a chain
// of 32 V_WMMA_F32_16X16X4_F32; diagonal of C = the 16 edge scores.

typedef __attribute__((ext_vector_type(2))) float v2f;
typedef __attribute__((ext_vector_type(8))) float v8f;

#define D_FEAT 128
#define EDGES_PER_WAVE 16
#define WAVES_PER_BLOCK 8
#define BLOCK_THREADS 256

__global__ __launch_bounds__(BLOCK_THREADS)
void edge_dot_wmma_f32(const float* __restrict__ h,
                       const long long* __restrict__ src,
                       const long long* __restrict__ dst,
                       float* __restrict__ out,
                       int n_edges) {
  const int lane = threadIdx.x & 31;
  const int wave = threadIdx.x >> 5;
  const int tile = blockIdx.x * WAVES_PER_BLOCK + wave;   // 16 edges per tile
  const int ebase = tile * EDGES_PER_WAVE;
  if (ebase >= n_edges) return;                           // wave-uniform skip

  const int col   = lane & 15;     // which of the 16 edges this lane serves
  const int khalf = lane >> 4;     // 0: K={kb,kb+1}, 1: K={kb+2,kb+3}

  // Clamp for partial tiles: keeps loads in-bounds while EXEC stays all-ones
  // (WMMA requires full EXEC). Stores are guarded separately.
  int e = ebase + col;
  if (e >= n_edges) e = n_edges - 1;

  // 64-bit edge indices from the reference; node ids fit in 32 bits.
  const int si = (int)src[e];
  const int di = (int)dst[e];
  const float* __restrict__ arow = h + (size_t)si * D_FEAT;  // hu row (A, M=col)
  const float* __restrict__ brow = h + (size_t)di * D_FEAT;  // hv row (B, N=col)

  v8f c = {0.f, 0.f, 0.f, 0.f, 0.f, 0.f, 0.f, 0.f};

  // 32 chained K=4 steps. A 16x4 f32 fragment per lane = 8 contiguous bytes of
  // its hu row; B 4x16 fragment per lane = 8 contiguous bytes of its hv row.
  // Each step: 2x global_load_b64 + 1x v_wmma_f32_16x16x4_f32.
#pragma unroll
  for (int kb = 0; kb < D_FEAT; kb += 4) {
    const int ko = kb + 2 * khalf;
    v2f a = *(const v2f*)(arow + ko);
    v2f b = *(const v2f*)(brow + ko);
    c = __builtin_amdgcn_wmma_f32_16x16x4_f32(
        /*neg_a=*/false, a, /*neg_b=*/false, b,
        /*c_mod=*/(short)0, c, /*reuse_a=*/false, /*reuse_b=*/false);
  }

  // Diagonal extraction from the 16x16 f32 C layout:
  //   i in 0..7  -> lane i,      VGPR i
  //   i in 8..15 -> lane i+16    (lanes 24..31), VGPR lane-24
  const bool lo = (lane < 8);
  const bool hi = (lane >= 24);
  if (lo || hi) {
    const int idx = lo ? lane : (lane - 24);         // which c[] element
    float val = c[0];
    if (idx == 1) val = c[1];
    if (idx == 2) val = c[2];
    if (idx == 3) val = c[3];
    if (idx == 4) val = c[4];
    if (idx == 5) val = c[5];
    if (idx == 6) val = c[6];
    if (idx == 7) val = c[7];
    const int eout = ebase + (lo ? lane : (lane - 16));  // diagonal index i
    if (eout < n_edges) out[eout] = val;
  }
}

extern "C" void kernel_launch(void* const* d_in, const int* in_sizes, int n_in,
                              void* d_out, int out_size, void* d_ws, size_t ws_size,
                              hipStream_t stream) {
  const float*     h   = (const float*)d_in[0];
  const long long* src = (const long long*)d_in[1];
  const long long* dst = (const long long*)d_in[2];
  float*           out = (float*)d_out;

  const int n_edges = in_sizes[1];                 // src has one entry per edge
  const int tiles   = (n_edges + EDGES_PER_WAVE - 1) / EDGES_PER_WAVE;
  const int blocks  = (tiles + WAVES_PER_BLOCK - 1) / WAVES_PER_BLOCK;

  edge_dot_wmma_f32<<<blocks, BLOCK_THREADS, 0, stream>>>(h, src, dst, out, n_edges);
}